// ModifiedGATModule_86088324481253
// MI455X (gfx1250) — compile-verified
//
#include <hip/hip_runtime.h>
#include <hip/hip_bf16.h>

typedef float v2f __attribute__((ext_vector_type(2)));
typedef float v8f __attribute__((ext_vector_type(8)));

// D = A(16x4) * B(4x16) + C, fp32, wave32.
__device__ __forceinline__ v8f wmma_f32(v2f a, v2f b, v8f c) {
  return __builtin_amdgcn_wmma_f32_16x16x4_f32(
      /*neg_a=*/false, a, /*neg_b=*/false, b,
      /*c_mod=*/(short)0, c, /*reuse_a=*/false, /*reuse_b=*/false);
}

#define GAT_N 4096

// h1[N,256] = x[N,128] @ W[128,256].  grid(256,4) x block(128): wave per 16x16 tile.
__global__ void __launch_bounds__(128) gat_gemm1(const float* __restrict__ x,
                                                 const float* __restrict__ W,
                                                 float* __restrict__ h1) {
  const int lane = threadIdx.x & 31;
  const int wave = threadIdx.x >> 5;
  const int lo = lane & 15, hi = lane >> 4;
  const int i0 = blockIdx.x * 16;
  const int n0 = blockIdx.y * 64 + wave * 16;
  v8f c = {};
  const float* xrow = x + (size_t)(i0 + lo) * 128;
  for (int k0 = 0; k0 < 128; k0 += 4) {
    v2f a = *(const v2f*)(xrow + k0 + 2 * hi);
    v2f b;
    b.x = W[(size_t)(k0 + 2 * hi + 0) * 256 + n0 + lo];
    b.y = W[(size_t)(k0 + 2 * hi + 1) * 256 + n0 + lo];
    c = wmma_f32(a, b, c);
  }
#pragma unroll
  for (int r = 0; r < 8; ++r)
    h1[(size_t)(i0 + r + 8 * hi) * 256 + n0 + lo] = c[r];
}

// h2[N,64] = elu(hin[N,256]) @ W[256,64].  grid(256) x block(128).
__global__ void __launch_bounds__(128) gat_gemm2_elu(const float* __restrict__ hin,
                                                     const float* __restrict__ W,
                                                     float* __restrict__ h2) {
  const int lane = threadIdx.x & 31;
  const int wave = threadIdx.x >> 5;
  const int lo = lane & 15, hi = lane >> 4;
  const int i0 = blockIdx.x * 16;
  const int n0 = wave * 16;
  v8f c = {};
  const float* arow = hin + (size_t)(i0 + lo) * 256;
  for (int k0 = 0; k0 < 256; k0 += 4) {
    v2f a = *(const v2f*)(arow + k0 + 2 * hi);
    a.x = a.x > 0.f ? a.x : (__expf(a.x) - 1.f);
    a.y = a.y > 0.f ? a.y : (__expf(a.y) - 1.f);
    v2f b;
    b.x = W[(size_t)(k0 + 2 * hi + 0) * 64 + n0 + lo];
    b.y = W[(size_t)(k0 + 2 * hi + 1) * 64 + n0 + lo];
    c = wmma_f32(a, b, c);
  }
#pragma unroll
  for (int r = 0; r < 8; ++r)
    h2[(size_t)(i0 + r + 8 * hi) * 64 + n0 + lo] = c[r];
}

// Transposed attention coefficients: es_t[h*N + n], ed_t[h*N + n] (contiguous per head).
__global__ void gat_edots(const float* __restrict__ h, const float* __restrict__ a_src,
                          const float* __restrict__ a_dst, float* __restrict__ es_t,
                          float* __restrict__ ed_t, int NH, int H, int F) {
  int t = blockIdx.x * blockDim.x + threadIdx.x;
  if (t >= NH) return;
  int n = t & (GAT_N - 1);
  int hh = t >> 12;  // t / GAT_N
  const float* hp = h + (size_t)n * H * F + hh * F;
  const float* as = a_src + hh * F;
  const float* ad = a_dst + hh * F;
  float s = 0.f, d = 0.f;
  for (int f = 0; f < F; ++f) {
    float v = hp[f];
    s += v * as[f];
    d += v * ad[f];
  }
  es_t[(size_t)hh * GAT_N + n] = s;
  ed_t[(size_t)hh * GAT_N + n] = d;
}

// Flash-style masked softmax + aggregation.
// Wave (wid) = (head h, j-segment seg). SPLIT>1 requires H==1 and 8 waves/block;
// partial (m,l,acc) per segment are merged with the online-softmax combine.
template <int H, int SPLIT, int F, int TJ>
__global__ void __launch_bounds__(32 * H * SPLIT)
gat_attn_agg(const float* __restrict__ es_t, const float* __restrict__ ed_t,
             const unsigned char* __restrict__ adj, const float* __restrict__ hmat,
             float* __restrict__ out) {
  constexpr int W = H * SPLIT;
  static_assert(TJ == 64 && F % 16 == 0, "tile shapes");
  static_assert(SPLIT == 1 || (H == 1 && W == 8), "split path needs H==1, 8 waves");
  constexpr int NF = F / 16;
  constexpr int PS = TJ + 4;  // stride 68: 16B-aligned rows, conflict-free A-reads
  __shared__ __align__(16) float p_s[W][16][PS];
  __shared__ float m_s[W][16];
  __shared__ float l_s[W][16];
  __shared__ float rs_s[W][16];
  __shared__ float M_s[16];
  __shared__ float L_s[16];

  const int lane = threadIdx.x & 31;
  const int wid = threadIdx.x >> 5;
  const int h = wid % H;
  const int seg = wid / H;
  const int lo = lane & 15, hi = lane >> 4;
  const int i0 = blockIdx.x * 16;
  const int row = lane >> 1;  // logit-phase row (2 lanes per row)
  const int sub = lane & 1;   // half of the j-tile this lane covers

  const float NEGBIG = -1e30f;
  if (lane < 16) {
    m_s[wid][lane] = NEGBIG;
    l_s[wid][lane] = 0.f;
  }
  v8f zero = {};
  v8f acc[NF];
#pragma unroll
  for (int nf = 0; nf < NF; ++nf) acc[nf] = zero;

  const float es = es_t[(size_t)h * GAT_N + i0 + row];
  const float* edh = ed_t + (size_t)h * GAT_N;
  const unsigned char* adjrow = adj + (size_t)(i0 + row) * GAT_N;

  const int jbeg = seg * (GAT_N / SPLIT);
  const int jend = jbeg + (GAT_N / SPLIT);

  for (int j0 = jbeg; j0 < jend; j0 += TJ) {
    const int jb = j0 + sub * (TJ / 2);
    if (j0 + TJ < jend) __builtin_prefetch(adjrow + jb + TJ, 0, 0);

    // ---- phase 1: vector-load adj bytes + e_dst, build masked logits in regs
    uint4 am0 = *(const uint4*)(adjrow + jb);
    uint4 am1 = *(const uint4*)(adjrow + jb + 16);
    unsigned int aw[8] = {am0.x, am0.y, am0.z, am0.w, am1.x, am1.y, am1.z, am1.w};
    float edf[32];
#pragma unroll
    for (int q = 0; q < 8; ++q) {
      float4 t4 = *(const float4*)(edh + jb + 4 * q);
      edf[4 * q + 0] = t4.x;
      edf[4 * q + 1] = t4.y;
      edf[4 * q + 2] = t4.z;
      edf[4 * q + 3] = t4.w;
    }
    float ev[32];
    float tmax = NEGBIG;
#pragma unroll
    for (int k = 0; k < 32; ++k) {
      bool edge = ((aw[k >> 2] >> (8 * (k & 3))) & 0xffu) != 0u;
      float v = es + edf[k];
      v = (v >= 0.f) ? v : 0.2f * v;  // leaky_relu
      float e = edge ? v : NEGBIG;
      ev[k] = e;
      tmax = fmaxf(tmax, e);
    }
    tmax = fmaxf(tmax, __shfl_xor(tmax, 1, 32));

    // ---- phase 2: online-softmax bookkeeping, exponentiate, write p-tile (b128s)
    float m_old = m_s[wid][row];
    float m_new = fmaxf(m_old, tmax);
    float rs = __expf(m_old - m_new);  // ==1 in the degenerate all-masked case
    float psum = 0.f;
    float pv[32];
#pragma unroll
    for (int k = 0; k < 32; ++k) {
      float p = __expf(ev[k] - m_new);
      pv[k] = p;
      psum += p;
    }
    float* prow = &p_s[wid][row][sub * (TJ / 2)];
#pragma unroll
    for (int q = 0; q < 8; ++q)
      *(float4*)(prow + 4 * q) =
          make_float4(pv[4 * q], pv[4 * q + 1], pv[4 * q + 2], pv[4 * q + 3]);
    psum += __shfl_xor(psum, 1, 32);
    if (sub == 0) {
      m_s[wid][row] = m_new;
      l_s[wid][row] = l_s[wid][row] * rs + psum;
      rs_s[wid][row] = rs;
    }

    // ---- phase 3: rescale accumulators (per C-fragment row)
#pragma unroll
    for (int r = 0; r < 8; ++r) {
      float rr = rs_s[wid][r + 8 * hi];
#pragma unroll
      for (int nf = 0; nf < NF; ++nf) acc[nf][r] *= rr;
    }

    // ---- phase 4: acc += P(16xTJ) @ Hmat(TJ x F) via f32 WMMA, K=4 steps
    for (int k0 = 0; k0 < TJ; k0 += 4) {
      v2f a = *(const v2f*)&p_s[wid][lo][k0 + 2 * hi];
      const float* hb = hmat + (size_t)(j0 + k0 + 2 * hi) * (H * F) + h * F;
#pragma unroll
      for (int nf = 0; nf < NF; ++nf) {
        v2f b;
        b.x = hb[nf * 16 + lo];
        b.y = hb[(size_t)H * F + nf * 16 + lo];
        acc[nf] = wmma_f32(a, b, acc[nf]);
      }
    }
  }

  if constexpr (SPLIT == 1) {
    // ---- epilogue: normalize and store (waves fully independent)
#pragma unroll
    for (int r = 0; r < 8; ++r) {
      float inv = 1.f / l_s[wid][r + 8 * hi];
#pragma unroll
      for (int nf = 0; nf < NF; ++nf)
        out[(size_t)(i0 + r + 8 * hi) * (H * F) + h * F + nf * 16 + lo] =
            acc[nf][r] * inv;
    }
  } else {
    // ---- cross-segment flash combine (H==1, 8 waves, 256 threads)
    __syncthreads();
    if (threadIdx.x < 16) {
      int rr = threadIdx.x;
      float M = NEGBIG;
#pragma unroll
      for (int w = 0; w < W; ++w) M = fmaxf(M, m_s[w][rr]);
      float L = 0.f;
#pragma unroll
      for (int w = 0; w < W; ++w) L += l_s[w][rr] * __expf(m_s[w][rr] - M);
      M_s[rr] = M;
      L_s[rr] = L;
    }
    __syncthreads();
    // rescale this wave's acc into its (now free) p-slab
#pragma unroll
    for (int r = 0; r < 8; ++r) {
      int rr = r + 8 * hi;
      float f = __expf(m_s[wid][rr] - M_s[rr]);
#pragma unroll
      for (int nf = 0; nf < NF; ++nf) p_s[wid][rr][nf * 16 + lo] = acc[nf][r] * f;
    }
    __syncthreads();
    // 256 threads reduce 16 x F slab over 8 segments, normalize, store
    int t = threadIdx.x;
    int rr = t >> 4;
    int c4 = (t & 15) * 4;
    float s0 = 0.f, s1 = 0.f, s2 = 0.f, s3 = 0.f;
#pragma unroll
    for (int w = 0; w < W; ++w) {
      float4 v = *(const float4*)&p_s[w][rr][c4];
      s0 += v.x;
      s1 += v.y;
      s2 += v.z;
      s3 += v.w;
    }
    float inv = 1.f / L_s[rr];
    float4 o = make_float4(s0 * inv, s1 * inv, s2 * inv, s3 * inv);
    *(float4*)&out[(size_t)(i0 + rr) * F + c4] = o;
  }
}

extern "C" void kernel_launch(void* const* d_in, const int* in_sizes, int n_in,
                              void* d_out, int out_size, void* d_ws, size_t ws_size,
                              hipStream_t stream) {
  (void)in_sizes; (void)n_in; (void)out_size; (void)ws_size;
  const float* x = (const float*)d_in[0];
  const unsigned char* adj = (const unsigned char*)d_in[1];  // jnp.bool_ -> 1 byte
  const float* W1 = (const float*)d_in[2];
  const float* a1s = (const float*)d_in[3];
  const float* a1d = (const float*)d_in[4];
  const float* W2 = (const float*)d_in[5];
  const float* a2s = (const float*)d_in[6];
  const float* a2d = (const float*)d_in[7];
  float* out = (float*)d_out;

  float* ws = (float*)d_ws;
  float* h1 = ws;                   // 4096*256
  float* h1agg = h1 + GAT_N * 256;  // 4096*256
  float* h2 = h1agg + GAT_N * 256;  // 4096*64
  float* es1 = h2 + GAT_N * 64;     // 8*4096 (transposed per head)
  float* ed1 = es1 + GAT_N * 8;     // 8*4096
  float* es2 = ed1 + GAT_N * 8;     // 4096
  float* ed2 = es2 + GAT_N;         // 4096

  // Layer 1: H=8, F=32 (one wave per head, 8 waves/block)
  gat_gemm1<<<dim3(GAT_N / 16, 4), 128, 0, stream>>>(x, W1, h1);
  gat_edots<<<(GAT_N * 8 + 255) / 256, 256, 0, stream>>>(h1, a1s, a1d, es1, ed1,
                                                         GAT_N * 8, 8, 32);
  gat_attn_agg<8, 1, 32, 64><<<GAT_N / 16, 256, 0, stream>>>(es1, ed1, adj, h1, h1agg);

  // Layer 2: H=1, F=64, j split across 8 waves + flash combine
  gat_gemm2_elu<<<GAT_N / 16, 128, 0, stream>>>(h1agg, W2, h2);
  gat_edots<<<(GAT_N + 255) / 256, 256, 0, stream>>>(h2, a2s, a2d, es2, ed2,
                                                     GAT_N, 1, 64);
  gat_attn_agg<1, 8, 64, 64><<<GAT_N / 16, 256, 0, stream>>>(es2, ed2, adj, h2, out);
}